// _EKF_Component_35510789603520
// MI455X (gfx1250) — compile-verified
//
#include <hip/hip_runtime.h>
#include <math.h>

// CDNA5 / gfx1250 EKF step: x_new = m - r*z,  where (F P F^T + (q+r)I) z = m - F x
// Heavy work (two 64x64x64 fp32 GEMMs per batch) runs on V_WMMA_F32_16X16X4_F32.

typedef __attribute__((ext_vector_type(2))) float v2f;
typedef __attribute__((ext_vector_type(8))) float v8f;

#define LSTR 66        // padded LDS row stride (floats) to break bank conflicts
#define NSTATE 64

__global__ __launch_bounds__(512) void ekf_step_kernel(
    const float* __restrict__ meas,    // [B,64] measurement_source
    const float* __restrict__ xk,      // [B,64] x_kalman
    const float* __restrict__ Pk,      // [B,64,64] P_kalman
    const float* __restrict__ Fm,      // [64,64] F
    const float* __restrict__ log_q,   // scalar
    const float* __restrict__ log_r,   // scalar
    float* __restrict__ xout)          // [B,64]
{
    __shared__ float sF[NSTATE * LSTR];   // F
    __shared__ float sP[NSTATE * LSTR];   // P, then reused as S = T*F^T + (q+r)I
    __shared__ float sT[NSTATE * LSTR];   // T = F*P
    __shared__ float sx[NSTATE];          // x_kalman
    __shared__ float sm[NSTATE];          // measurement
    __shared__ float sb[NSTATE];          // innov / RHS working vector

    const int b = blockIdx.x;
    const int t = threadIdx.x;
    const float q = expf(log_q[0]);
    const float r = expf(log_r[0]);
    const float qr = q + r;

    // ---- stage F and P into LDS (coalesced) ----
    for (int i = t; i < NSTATE * NSTATE; i += 512) {
        const int row = i >> 6, col = i & 63;
        sF[row * LSTR + col] = Fm[i];
        sP[row * LSTR + col] = Pk[(size_t)b * 4096 + i];
    }
    if (t < NSTATE) {
        sx[t] = xk[(size_t)b * NSTATE + t];
        sm[t] = meas[(size_t)b * NSTATE + t];
    }
    __syncthreads();

    // ---- wave/tile decomposition: 16 waves -> 4x4 tiles of the 64x64 result ----
    const int wave = t >> 5;
    const int lane = t & 31;
    const int ti = wave >> 2;          // tile row  (0..3)
    const int tj = wave & 3;           // tile col  (0..3)
    const int l  = lane & 15;          // lane within half
    const int h  = lane >> 4;          // half select
    const int arow = 16 * ti + l;      // A-fragment row (M)
    const int bcol = 16 * tj + l;      // B-fragment col (N)

    // ---- GEMM1: T = F * P  (16 chunks of K=4) ----
    v8f acc = {0.f, 0.f, 0.f, 0.f, 0.f, 0.f, 0.f, 0.f};
#pragma unroll
    for (int c = 0; c < 16; ++c) {
        const int k0 = 4 * c + 2 * h;  // lanes 0-15: K=4c,4c+1 ; lanes 16-31: K=4c+2,4c+3
        v2f a, bb;
        a[0]  = sF[arow * LSTR + k0];
        a[1]  = sF[arow * LSTR + k0 + 1];
        bb[0] = sP[k0 * LSTR + bcol];
        bb[1] = sP[(k0 + 1) * LSTR + bcol];
        acc = __builtin_amdgcn_wmma_f32_16x16x4_f32(
            /*neg_a=*/false, a, /*neg_b=*/false, bb,
            /*c_mod=*/(short)0, acc, /*reuse_a=*/false, /*reuse_b=*/false);
    }
#pragma unroll
    for (int v = 0; v < 8; ++v)
        sT[(16 * ti + v + 8 * h) * LSTR + 16 * tj + l] = acc[v];
    __syncthreads();

    // ---- GEMM2: S = T * F^T + (q+r) I   (written over sP) ----
    v8f acc2 = {0.f, 0.f, 0.f, 0.f, 0.f, 0.f, 0.f, 0.f};
#pragma unroll
    for (int c = 0; c < 16; ++c) {
        const int k0 = 4 * c + 2 * h;
        v2f a, bb;
        a[0]  = sT[arow * LSTR + k0];
        a[1]  = sT[arow * LSTR + k0 + 1];
        bb[0] = sF[bcol * LSTR + k0];      // F^T[k][n] = F[n][k]
        bb[1] = sF[bcol * LSTR + k0 + 1];
        acc2 = __builtin_amdgcn_wmma_f32_16x16x4_f32(
            false, a, false, bb, (short)0, acc2, false, false);
    }
#pragma unroll
    for (int v = 0; v < 8; ++v) {
        const int row = 16 * ti + v + 8 * h;
        const int col = 16 * tj + l;
        sP[row * LSTR + col] = acc2[v] + ((row == col) ? qr : 0.f);
    }
    __syncthreads();

    // ---- innovation: b = m - F x  (x_pred GEMV, 64 threads) ----
    if (t < NSTATE) {
        float xp = 0.f;
#pragma unroll 8
        for (int k = 0; k < NSTATE; ++k)
            xp += sF[t * LSTR + k] * sx[k];
        sb[t] = sm[t] - xp;
    }
    __syncthreads();

    // ---- Gauss-Jordan solve: S z = b  (S is SPD + r-shifted; no pivoting) ----
    // 8 threads per row (all within one wave -> lockstep read of the factor
    // column before its own writes); one barrier per pivot step.
    const int row = t >> 3;            // 0..63
    const int cth = t & 7;             // 0..7 column strider
    for (int p = 0; p < NSTATE; ++p) {
        const float pinv = 1.0f / sP[p * LSTR + p];
        if (row != p) {
            const float factor = sP[row * LSTR + p] * pinv;
            for (int j = p + cth; j < NSTATE; j += 8)
                sP[row * LSTR + j] -= factor * sP[p * LSTR + j];
            if (cth == 0)
                sb[row] -= factor * sb[p];
        }
        __syncthreads();
    }

    // ---- x_new = m - r * z,  z = b / diag(S) ----
    if (t < NSTATE) {
        const float z = sb[t] / sP[t * LSTR + t];
        xout[(size_t)b * NSTATE + t] = sm[t] - r * z;
    }
}

extern "C" void kernel_launch(void* const* d_in, const int* in_sizes, int n_in,
                              void* d_out, int out_size, void* d_ws, size_t ws_size,
                              hipStream_t stream) {
    // setup_inputs order: y, measurement_source, x_kalman, P_kalman, F, log_q, log_r
    const float* y    = (const float*)d_in[0]; (void)y;   // unused by the math
    const float* meas = (const float*)d_in[1];
    const float* xk   = (const float*)d_in[2];
    const float* Pk   = (const float*)d_in[3];
    const float* Fm   = (const float*)d_in[4];
    const float* lq   = (const float*)d_in[5];
    const float* lr   = (const float*)d_in[6];
    float* out = (float*)d_out;

    const int B = in_sizes[1] / NSTATE;   // 8192
    ekf_step_kernel<<<dim3(B), dim3(512), 0, stream>>>(meas, xk, Pk, Fm, lq, lr, out);
}